// DTW_40312563041028
// MI455X (gfx1250) — compile-verified
//
#include <hip/hip_runtime.h>
#include <math.h>

// Problem constants (match reference)
#define BATCH 32
#define NN 1024
#define MM 1024
#define FF 64
#define BAND 16
#define NTHREADS 1024
#define NWAVES 32

typedef __attribute__((ext_vector_type(2))) float v2f;
typedef __attribute__((ext_vector_type(8))) float v8f;

__global__ __launch_bounds__(NTHREADS)
void dtw_fused_kernel(const float* __restrict__ x,
                      const float* __restrict__ y,
                      float* __restrict__ out) {
    __shared__ float s_band[BAND * MM];   // 64 KB cost band
    __shared__ float s_xsq[NN];
    __shared__ float s_ysq[MM];
    __shared__ float s_scanA[NWAVES];     // per-wave cumsum totals
    __shared__ float s_scanB[NWAVES];     // per-wave cummin totals
    __shared__ float s_edge[NWAVES];      // wave-boundary D[i-1] values

    const int b    = blockIdx.x;
    const int tid  = threadIdx.x;
    const int lane = tid & 31;
    const int wid  = tid >> 5;

    const float* xb = x + (size_t)b * NN * FF;
    const float* yb = y + (size_t)b * MM * FF;

    // ---- squared norms: thread tid owns row tid of x and row tid of y ----
    {
        const float4* xr4 = (const float4*)(xb + (size_t)tid * FF);
        const float4* yr4 = (const float4*)(yb + (size_t)tid * FF);
        float sx = 0.f, sy = 0.f;
        #pragma unroll
        for (int k = 0; k < FF / 4; ++k) {
            float4 a = xr4[k];
            float4 c = yr4[k];
            sx += a.x * a.x + a.y * a.y + a.z * a.z + a.w * a.w;
            sy += c.x * c.x + c.y * c.y + c.z * c.z + c.w * c.w;
        }
        s_xsq[tid] = sx;
        s_ysq[tid] = sy;
    }
    __syncthreads();

    // WMMA f32 16x16x4 fragment mapping (ISA: 32-bit A 16x4 layout):
    // lanes 0-15 hold K = k0,k0+1 for row (lane&15); lanes 16-31 hold K = k0+2,k0+3.
    const int kbase = (lane >> 4) * 2;
    const int mrow  = lane & 15;
    const int hi    = lane >> 4;

    float result   = 0.0f;
    float cur_prev = 0.0f;                 // D[i-1, tid], carried in registers
    const float INF = __builtin_inff();

    for (int band = 0; band < NN / BAND; ++band) {
        const int r0 = band * BAND;

        // -------- Phase 1: WMMA cost band (16 rows x 1024 cols) --------
        v2f af[16];
        const float* xr = xb + (size_t)(r0 + mrow) * FF;
        #pragma unroll
        for (int k = 0; k < 16; ++k)
            af[k] = *(const v2f*)(xr + 4 * k + kbase);

        // Prefetch next band's A rows; hidden behind this band's GEMM + DP.
        if (band + 1 < NN / BAND)
            __builtin_prefetch(xb + (size_t)(r0 + BAND + mrow) * FF, 0, 0);

        // Rolled tile loop: only one tile's B fragments live -> no spills.
        #pragma unroll 1
        for (int t = 0; t < 2; ++t) {
            const int n0 = (wid * 2 + t) * 16;       // column tile base
            const float* yr = yb + (size_t)(n0 + mrow) * FF;
            v8f acc = {};
            #pragma unroll
            for (int k = 0; k < 16; ++k) {
                v2f bk = *(const v2f*)(yr + 4 * k + kbase);
                acc = __builtin_amdgcn_wmma_f32_16x16x4_f32(
                    false, af[k], false, bk, (short)0, acc, false, false);
            }
            // C/D layout: VGPR v holds rows v (lanes 0-15) / v+8 (lanes 16-31).
            const int col = n0 + mrow;
            const float ysq = s_ysq[col];
            #pragma unroll
            for (int v = 0; v < 8; ++v) {
                const int lr = v + hi * 8;
                float sq = s_xsq[r0 + lr] + ysq - 2.0f * acc[v];
                s_band[lr * MM + col] = sqrtf(fmaxf(sq, 0.0f));
            }
        }
        __syncthreads();

        // -------- Phase 2: DP rows, 2 barriers per row ----------------------
        // cur = C + cummin(cost + mins - C), C = cumsum(cost).
        // Previous row lives in registers (cur_prev); only the 32 wave-boundary
        // values round-trip through s_edge. Cross-wave scan offsets are computed
        // redundantly by every wave (no dedicated top-scan barrier).
        for (int lr = 0; lr < BAND; ++lr) {
            const int i = r0 + lr;
            const float cost = s_band[lr * MM + tid];

            float mins;
            if (i == 0) {
                mins = (tid == 0) ? 0.0f : INF;
            } else {
                float diag = __shfl_up(cur_prev, 1, 32);
                if (lane == 0)
                    diag = (wid == 0) ? INF : s_edge[wid - 1];
                mins = fminf(cur_prev, diag);
            }

            // Intra-wave inclusive cumsum of cost.
            float Cw = cost;
            #pragma unroll
            for (int d = 1; d < 32; d <<= 1) {
                float n = __shfl_up(Cw, d, 32);
                if (lane >= d) Cw += n;
            }
            // Wave-relative t; global t = tw - off_w (off_w wave-uniform),
            // so the intra-wave cummin commutes with the offset.
            float tw = cost + mins - Cw;
            float mw = tw;
            #pragma unroll
            for (int d = 1; d < 32; d <<= 1) {
                float n = __shfl_up(mw, d, 32);
                if (lane >= d) mw = fminf(mw, n);
            }
            if (lane == 31) { s_scanA[wid] = Cw; s_scanB[wid] = mw; }
            __syncthreads();                               // barrier 1

            // Redundant cross-wave scan in every wave: lane l holds wave l's
            // partials; broadcast this wave's offset/carry via shfl(.., wid).
            {
                float p = s_scanA[lane];
                float q = s_scanB[lane];
                float s = p;
                #pragma unroll
                for (int d = 1; d < 32; d <<= 1) {
                    float n = __shfl_up(s, d, 32);
                    if (lane >= d) s += n;
                }
                float offl = __shfl_up(s, 1, 32);
                if (lane == 0) offl = 0.0f;                // exclusive sum
                float z = q - offl;
                float zz = z;
                #pragma unroll
                for (int d = 1; d < 32; d <<= 1) {
                    float n = __shfl_up(zz, d, 32);
                    if (lane >= d) zz = fminf(zz, n);
                }
                float el = __shfl_up(zz, 1, 32);
                if (lane == 0) el = INF;                   // exclusive min

                const float off = __shfl(offl, wid, 32);
                const float e   = __shfl(el,   wid, 32);
                const float C   = Cw + off;
                const float m   = fminf(e, mw - off);
                cur_prev = C + m;                          // D[i, tid]
            }
            if (lane == 31) s_edge[wid] = cur_prev;
            __syncthreads();                               // barrier 2

            if (i == NN - 1 && tid == MM - 1) result = cur_prev;
        }
    }

    if (tid == MM - 1) out[b] = result;
}

extern "C" void kernel_launch(void* const* d_in, const int* in_sizes, int n_in,
                              void* d_out, int out_size, void* d_ws, size_t ws_size,
                              hipStream_t stream) {
    (void)in_sizes; (void)n_in; (void)d_ws; (void)ws_size; (void)out_size;
    const float* x = (const float*)d_in[0];
    const float* y = (const float*)d_in[1];
    float* out = (float*)d_out;
    dtw_fused_kernel<<<BATCH, NTHREADS, 0, stream>>>(x, y, out);
}